// FocalLoss_20581483282497
// MI455X (gfx1250) — compile-verified
//
#include <hip/hip_runtime.h>

// ---------------------------------------------------------------------------
// FocalLoss (RetinaNet) for MI455X / gfx1250.
// Memory-bound: 307MB classification stream dominates -> ~14us at 23.3TB/s.
// Strategy: fused per-anchor matching (phase A), flat coalesced float4 focal
// stream (phase B), and a wave-per-image finalize whose big partial-sum
// reduction runs on V_WMMA_F32_16X16X4_F32 (B = ones => row sums, 64 f32
// reduced per instruction, chained through the C accumulator).
// ---------------------------------------------------------------------------

typedef __attribute__((ext_vector_type(2))) float v2f;
typedef __attribute__((ext_vector_type(8))) float v8f;

#define NN      120000
#define MM      32
#define KK      80
#define BB      8
#define THREADS 256
#define BLKA    469      // ceil(NN/256)
#define BLKB    1024     // blocks per image for the cls stream (16 wmma chunks)

__device__ __forceinline__ float clip01(float p) {
    return fminf(fmaxf(p, 1e-4f), 1.0f - 1e-4f);
}
// focal term for target==0:  0.75 * p^2 * (-log(1-p))
__device__ __forceinline__ float negloss(float p0) {
    float p = clip01(p0);
    return 0.75f * p * p * (-__logf(1.0f - p));
}
// focal term for target==1:  0.25 * (1-p)^2 * (-log p)
__device__ __forceinline__ float posloss(float p0) {
    float p = clip01(p0);
    float q = 1.0f - p;
    return 0.25f * q * q * (-__logf(p));
}

// ---------------------------------------------------------------------------
// Phase A: per-anchor IoU argmax vs M=32 boxes (LDS-cached), emits 16-bit
// meta (state | cls_idx<<2), instance focal loss, reg smooth-L1, counts.
// ---------------------------------------------------------------------------
__global__ __launch_bounds__(THREADS)
void phaseA_kernel(const float* __restrict__ inst,
                   const float4* __restrict__ reg4,
                   const float4* __restrict__ anc4,
                   const float* __restrict__ ann,
                   unsigned short* __restrict__ meta,
                   float4* __restrict__ partsA)
{
    __shared__ float  annS[MM * 6];
    __shared__ float4 red[THREADS];
    const int b   = blockIdx.y;
    const int tid = threadIdx.x;
    const int n   = blockIdx.x * THREADS + tid;

    if (tid < MM * 6) annS[tid] = ann[b * MM * 6 + tid];
    __syncthreads();

    float instL = 0.0f, regL = 0.0f, posf = 0.0f, posif = 0.0f;

    if (n < NN) {
        const float4 a   = anc4[n];
        const float  aw  = a.z - a.x, ah = a.w - a.y;
        const float  acx = a.x + 0.5f * aw, acy = a.y + 0.5f * ah;
        const float  aarea = aw * ah;

        float best = -3.4e38f; int bi = 0;
        #pragma unroll 4
        for (int m = 0; m < MM; ++m) {
            const float bx0 = annS[m*6+0], by0 = annS[m*6+1];
            const float bx1 = annS[m*6+2], by1 = annS[m*6+3];
            float iw = fmaxf(fminf(a.z, bx1) - fmaxf(a.x, bx0), 0.0f);
            float ih = fmaxf(fminf(a.w, by1) - fmaxf(a.y, by0), 0.0f);
            float inter = iw * ih;
            float ua  = fmaxf(aarea + (bx1-bx0)*(by1-by0) - inter, 1e-8f);
            float iou = inter / ua;
            if (annS[m*6+4] == -1.0f) iou = -1.0f;   // invalid annotation mask
            if (iou > best) { best = iou; bi = m; }  // first-max == jnp.argmax
        }
        const float g0 = annS[bi*6+0], g1 = annS[bi*6+1];
        const float g2 = annS[bi*6+2], g3 = annS[bi*6+3];
        const float g4 = annS[bi*6+4], g5 = annS[bi*6+5];

        const bool pos  = best >= 0.5f;
        const bool posi = best >= 0.3f;
        const bool neg  = best <  0.4f;
        const int  state = pos ? 2 : (neg ? 0 : 1);
        const int  ci    = (int)g4;
        meta[(size_t)b * NN + n] = (unsigned short)(state | (ci << 2));
        posf  = pos  ? 1.0f : 0.0f;
        posif = posi ? 1.0f : 0.0f;

        // instance focal loss (target 1/0 by flag if pos_i, else ignored)
        const float pi = inst[(size_t)b * NN + n];
        if (posi) {
            if      (g5 == 1.0f) instL = posloss(pi);
            else if (g5 == 0.0f) instL = negloss(pi);
        }

        // regression smooth-L1 (only pos anchors)
        if (pos) {
            const float4 r  = reg4[(size_t)b * NN + n];
            const float gwr = g2 - g0, ghr = g3 - g1;
            const float gcx = g0 + 0.5f * gwr, gcy = g1 + 0.5f * ghr;
            const float gw  = fmaxf(gwr, 1.0f), gh = fmaxf(ghr, 1.0f);
            const float t0  = ((gcx - acx) / aw) * 10.0f;   // /0.1
            const float t1  = ((gcy - acy) / ah) * 10.0f;
            const float t2  = __logf(gw / aw) * 5.0f;       // /0.2
            const float t3  = __logf(gh / ah) * 5.0f;
            const float c   = 1.0f / 9.0f;
            float d, s = 0.0f;
            d = fabsf(t0 - r.x); s += (d <= c) ? 4.5f*d*d : d - 0.5f*c;
            d = fabsf(t1 - r.y); s += (d <= c) ? 4.5f*d*d : d - 0.5f*c;
            d = fabsf(t2 - r.z); s += (d <= c) ? 4.5f*d*d : d - 0.5f*c;
            d = fabsf(t3 - r.w); s += (d <= c) ? 4.5f*d*d : d - 0.5f*c;
            regL = s;
        }
    }

    red[tid] = make_float4(instL, regL, posf, posif);
    __syncthreads();
    for (int s = THREADS / 2; s > 0; s >>= 1) {
        if (tid < s) {
            float4 x = red[tid], y = red[tid + s];
            red[tid] = make_float4(x.x+y.x, x.y+y.y, x.z+y.z, x.w+y.w);
        }
        __syncthreads();
    }
    if (tid == 0) partsA[b * BLKA + blockIdx.x] = red[0];
}

// ---------------------------------------------------------------------------
// Phase B: fully coalesced float4 stream over [N*K] classification focal.
// Each float4 lies inside one anchor row (K=80, 4|80). "Ignore" anchors skip
// the global load entirely (bandwidth win).
// ---------------------------------------------------------------------------
__global__ __launch_bounds__(THREADS)
void phaseB_kernel(const float4* __restrict__ cls4,
                   const unsigned short* __restrict__ meta,
                   float* __restrict__ partsB)
{
    const int b   = blockIdx.y;
    const int tid = threadIdx.x;
    const int total4 = NN * KK / 4;                 // 2,400,000 per image
    const float4* c4 = cls4 + (size_t)b * total4;
    const unsigned short* mt = meta + (size_t)b * NN;

    float acc = 0.0f;
    for (int i = blockIdx.x * THREADS + tid; i < total4; i += BLKB * THREADS) {
        const int n = i / (KK / 4);                 // /20 -> anchor index
        const int m = mt[n];
        const int state = m & 3;
        if (state == 1) continue;                   // ignore band: loss==0, skip load
        const float4 v = c4[i];
        if (state == 2) {                           // positive: one-hot target
            const int ci = m >> 2;
            const int kb = (i - n * (KK / 4)) * 4;
            acc += (kb + 0 == ci) ? posloss(v.x) : negloss(v.x);
            acc += (kb + 1 == ci) ? posloss(v.y) : negloss(v.y);
            acc += (kb + 2 == ci) ? posloss(v.z) : negloss(v.z);
            acc += (kb + 3 == ci) ? posloss(v.w) : negloss(v.w);
        } else {                                    // negative: target 0 everywhere
            acc += negloss(v.x) + negloss(v.y) + negloss(v.z) + negloss(v.w);
        }
    }

    __shared__ float red[THREADS];
    red[tid] = acc;
    __syncthreads();
    for (int s = THREADS / 2; s > 0; s >>= 1) {
        if (tid < s) red[tid] += red[tid + s];
        __syncthreads();
    }
    if (tid == 0) partsB[b * BLKB + blockIdx.x] = red[0];
}

// ---------------------------------------------------------------------------
// Finalize: 1 block, 8 waves, wave b owns image b. The 1024 cls partials per
// image are reduced with chained V_WMMA_F32_16X16X4_F32 (A = 64 partials,
// B = ones => D rows are partial row-sums, accumulated through C). One
// shfl_xor(16) folds the two row halves. Phase-A partials via shfl tree.
// ---------------------------------------------------------------------------
__global__ __launch_bounds__(256)
void finalize_kernel(const float* __restrict__ partsB,
                     const float4* __restrict__ partsA,
                     float* __restrict__ out)
{
    const int tid  = threadIdx.x;
    const int b    = tid >> 5;    // wave id == image id (8 waves of 32)
    const int lane = tid & 31;

    // --- cls sum via WMMA-chained reduction (exactly 16 chunks of 64) ---
    v8f acc = {0.f, 0.f, 0.f, 0.f, 0.f, 0.f, 0.f, 0.f};
    v2f ones; ones.x = 1.0f; ones.y = 1.0f;
    const float* pb = partsB + b * BLKB;
    #pragma unroll
    for (int c = 0; c < BLKB / 64; ++c) {
        v2f a;
        a.x = pb[c * 64 + lane];
        a.y = pb[c * 64 + 32 + lane];
        acc = __builtin_amdgcn_wmma_f32_16x16x4_f32(
            /*neg_a=*/false, a, /*neg_b=*/false, ones,
            /*c_mod=*/(short)0, acc, /*reuse_a=*/false, /*reuse_b=*/false);
    }
    float cs = acc[0] + acc[1] + acc[2] + acc[3] + acc[4] + acc[5] + acc[6] + acc[7];
    cs += __shfl_xor(cs, 16, 32);   // rows 0-7 (+cols) live in lanes 0-15, rows 8-15 in 16-31

    // --- phase A partials: inst, reg, npos, npos_i ---
    float i_s = 0.f, r_s = 0.f, p_s = 0.f, pi_s = 0.f;
    for (int j = lane; j < BLKA; j += 32) {
        const float4 q = partsA[b * BLKA + j];
        i_s += q.x; r_s += q.y; p_s += q.z; pi_s += q.w;
    }
    for (int d = 16; d >= 1; d >>= 1) {
        i_s  += __shfl_xor(i_s,  d, 32);
        r_s  += __shfl_xor(r_s,  d, 32);
        p_s  += __shfl_xor(p_s,  d, 32);
        pi_s += __shfl_xor(pi_s, d, 32);
    }

    __shared__ float sm[BB * 3];
    if (lane == 0) {
        const float npos  = fmaxf(p_s, 1.0f);
        const float nposi = fmaxf(pi_s, 1.0f);
        sm[b * 3 + 0] = i_s / nposi;                              // inst loss
        sm[b * 3 + 1] = cs / npos;                                // cls loss
        sm[b * 3 + 2] = (p_s > 0.0f) ? r_s / (4.0f * npos) : 0.f; // reg loss
    }
    __syncthreads();
    if (tid < 3) {
        float s = 0.0f;
        for (int q = 0; q < BB; ++q) s += sm[q * 3 + tid];
        out[tid] = s / (float)BB;                  // (il, cl, rl) means
    }
}

// ---------------------------------------------------------------------------
extern "C" void kernel_launch(void* const* d_in, const int* in_sizes, int n_in,
                              void* d_out, int out_size, void* d_ws, size_t ws_size,
                              hipStream_t stream)
{
    const float* inst = (const float*)d_in[0];   // [B,N,1]
    const float* cls  = (const float*)d_in[1];   // [B,N,K]
    const float* reg  = (const float*)d_in[2];   // [B,N,4]
    const float* anc  = (const float*)d_in[3];   // [1,N,4]
    const float* ann  = (const float*)d_in[4];   // [B,M,6]
    (void)in_sizes; (void)n_in; (void)out_size; (void)ws_size;

    char* ws = (char*)d_ws;
    // layout: meta (B*N u16 = 1,920,000 B, 16B-aligned regions follow)
    unsigned short* meta   = (unsigned short*)ws;
    float4*         partsA = (float4*)(ws + 1920000);            // B*BLKA float4 = 60,032 B
    float*          partsB = (float*)(ws + 1920000 + 60032);     // B*BLKB float  = 32,768 B

    phaseA_kernel<<<dim3(BLKA, BB), THREADS, 0, stream>>>(
        inst, (const float4*)reg, (const float4*)anc, ann, meta, partsA);
    phaseB_kernel<<<dim3(BLKB, BB), THREADS, 0, stream>>>(
        (const float4*)cls, meta, partsB);
    finalize_kernel<<<1, 256, 0, stream>>>(partsB, (const float4*)partsA,
                                           (float*)d_out);
}